// RotatedMultiBoxLoss_38551626449491
// MI455X (gfx1250) — compile-verified
//
#include <hip/hip_runtime.h>
#include <stdint.h>

// Problem constants from the reference: B, T, P, C = (16, 50, 65536, 21)
constexpr int   kB   = 16;
constexpr int   kT   = 50;
constexpr int   kP   = 65536;
constexpr float kThr = 0.5f;
constexpr int   kBlk = 256;   // 8 wave32 waves per block
constexpr int   kPRR = 4;     // priors per thread in phase 1

// ---------------------------------------------------------------------------
// Phase 0: zero the packed (iou_bits << 32 | ~prior_idx) argmax cells.
// Must run every call (harness does not re-poison between replays).
// ---------------------------------------------------------------------------
__global__ __launch_bounds__(kBlk)
void ws_init_kernel(unsigned long long* __restrict__ ws) {
  int i = blockIdx.x * blockDim.x + threadIdx.x;
  if (i < kB * kT) ws[i] = 0ull;
}

// ---------------------------------------------------------------------------
// Phase 1: per (batch, target) argmax over all priors of axis-aligned IoU.
// Packed-u64 atomic max: IoU >= 0 so its float bits are monotone; ~idx in the
// low word makes the smallest prior index win ties (matches jnp.argmax).
// ---------------------------------------------------------------------------
__global__ __launch_bounds__(kBlk)
void best_prior_kernel(const float* __restrict__ priors,
                       const float* __restrict__ targets,
                       unsigned long long* __restrict__ ws) {
  const int b   = blockIdx.y;
  const int tid = threadIdx.x;
  const int p0  = blockIdx.x * (kBlk * kPRR);

  __shared__ float s_t[kT * 5];  // target point-form: ltx,lty,rbx,rby,area
  if (tid < kT) {
    const float* tg = targets + ((size_t)b * kT + tid) * 6;
    float cx = tg[0], cy = tg[1], w = tg[2], h = tg[3];
    s_t[tid * 5 + 0] = cx - 0.5f * w;
    s_t[tid * 5 + 1] = cy - 0.5f * h;
    s_t[tid * 5 + 2] = cx + 0.5f * w;
    s_t[tid * 5 + 3] = cy + 0.5f * h;
    s_t[tid * 5 + 4] = w * h;
  }
  __syncthreads();

  // Cache this thread's 4 priors in point-form (registers).
  float lx[kPRR], ly[kPRR], rx[kPRR], ry[kPRR], ar[kPRR];
#pragma unroll
  for (int k = 0; k < kPRR; ++k) {
    int p = p0 + k * kBlk + tid;               // coalesced across lanes
    const float* pr = priors + (size_t)p * 5;
    float cx = pr[0], cy = pr[1], w = pr[2], h = pr[3];
    lx[k] = cx - 0.5f * w;  ly[k] = cy - 0.5f * h;
    rx[k] = cx + 0.5f * w;  ry[k] = cy + 0.5f * h;
    ar[k] = w * h;
  }

  for (int t = 0; t < kT; ++t) {
    float tlx = s_t[t * 5 + 0], tly = s_t[t * 5 + 1];
    float trx = s_t[t * 5 + 2], trY = s_t[t * 5 + 3];
    float tar = s_t[t * 5 + 4];

    float bestv = -1.0f;  // IoU >= 0, so k==0 always replaces this
    int   bestp = 0;
#pragma unroll
    for (int k = 0; k < kPRR; ++k) {
      float ix = fminf(rx[k], trx) - fmaxf(lx[k], tlx);
      float iy = fminf(ry[k], trY) - fmaxf(ly[k], tly);
      ix = fmaxf(ix, 0.0f); iy = fmaxf(iy, 0.0f);
      float inter = ix * iy;
      float iou   = inter / (ar[k] + tar - inter);
      if (iou > bestv) { bestv = iou; bestp = p0 + k * kBlk + tid; }
    }

    unsigned long long pk =
        ((unsigned long long)__float_as_uint(bestv) << 32) |
        (unsigned long long)(unsigned)(~(unsigned)bestp);

    // wave32 max-reduce of the packed key, then one atomic per wave.
#pragma unroll
    for (int off = 16; off > 0; off >>= 1) {
      unsigned long long o = __shfl_xor(pk, off, 32);
      if (o > pk) pk = o;
    }
    if ((tid & 31) == 0) atomicMax(&ws[(size_t)b * kT + t], pk);
  }
}

// ---------------------------------------------------------------------------
// Phase 2: per prior -> best target (max over 50 IoUs), apply best-prior
// overrides, gather matched box, encode, emit conf. Targets are DMA'd to LDS
// with CDNA5 async global->LDS loads (ASYNCcnt) overlapped with decoding the
// packed argmax words.
// ---------------------------------------------------------------------------
__global__ __launch_bounds__(kBlk)
void match_encode_kernel(const float* __restrict__ priors,
                         const float* __restrict__ targets,
                         const unsigned long long* __restrict__ ws,
                         float* __restrict__ out_loc,
                         float* __restrict__ out_conf) {
  const int b   = blockIdx.y;
  const int tid = threadIdx.x;
  const int p   = blockIdx.x * kBlk + tid;

  __shared__ float s_tg[kT * 6];   // raw targets for this batch (300 floats)
  __shared__ int   s_bpi[kT];      // best prior index per target

  // Async DMA: 150 lanes x 8B = 1200 B of targets -> LDS (GLOBAL_LOAD_ASYNC_TO_LDS_B64).
  if (tid < (kT * 6) / 2) {
    unsigned lds_off = (unsigned)(size_t)&s_tg[tid * 2];  // low 32 bits of flat = LDS offset
    unsigned voff    = (unsigned)(tid * 8);
    unsigned long long base =
        (unsigned long long)(size_t)(targets + (size_t)b * kT * 6);
    asm volatile("global_load_async_to_lds_b64 %0, %1, %2 offset:0"
                 :
                 : "v"(lds_off), "v"(voff), "s"(base)
                 : "memory");
  }

  // Overlap: prefetch this thread's prior row and decode packed argmax words.
  __builtin_prefetch(priors + (size_t)p * 5, 0, 0);
  if (tid < kT) s_bpi[tid] = (int)~(unsigned)ws[(size_t)b * kT + tid];

  asm volatile("s_wait_asynccnt 0" ::: "memory");
  __syncthreads();

  // Prior: center form + point form + area.
  const float* pr = priors + (size_t)p * 5;
  float pcx = pr[0], pcy = pr[1], pw = pr[2], ph = pr[3], pang = pr[4];
  float plx = pcx - 0.5f * pw, ply = pcy - 0.5f * ph;
  float prx = pcx + 0.5f * pw, pry = pcy + 0.5f * ph;
  float parea = pw * ph;

  // Max/argmax over targets (ascending t + strict '>' == first-index argmax).
  float bestv = -1.0f;
  int   bestt = 0;
  for (int t = 0; t < kT; ++t) {
    float tcx = s_tg[t * 6 + 0], tcy = s_tg[t * 6 + 1];
    float tw  = s_tg[t * 6 + 2], th  = s_tg[t * 6 + 3];
    float tlx = tcx - 0.5f * tw, tly = tcy - 0.5f * th;
    float trx = tcx + 0.5f * tw, trY = tcy + 0.5f * th;
    float ix = fminf(prx, trx) - fmaxf(plx, tlx);
    float iy = fminf(pry, trY) - fmaxf(ply, tly);
    ix = fmaxf(ix, 0.0f); iy = fmaxf(iy, 0.0f);
    float inter = ix * iy;
    float iou   = inter / (parea + tw * th - inter);
    if (iou > bestv) { bestv = iou; bestt = t; }
  }

  // Overrides: each target claims its best prior; later t wins (scatter order).
  for (int t = 0; t < kT; ++t) {
    if (s_bpi[t] == p) { bestv = 2.0f; bestt = t; }
  }

  const float* mt = &s_tg[bestt * 6];
  float mcx = mt[0], mcy = mt[1], mw = mt[2], mh = mt[3], mang = mt[4];
  int conf = (bestv < kThr) ? 0 : ((int)mt[5] + 1);

  // encode(): VAR_CXCY=0.1, VAR_WH=0.2, VAR_ANG=0.1
  float l0 = (mcx - pcx) / (0.1f * pw);
  float l1 = (mcy - pcy) / (0.1f * ph);
  float l2 = __logf(mw / pw) * 5.0f;
  float l3 = __logf(mh / ph) * 5.0f;
  float l4 = (mang - pang) * 10.0f;

  size_t o = ((size_t)b * kP + p) * 5;
  out_loc[o + 0] = l0;
  out_loc[o + 1] = l1;
  out_loc[o + 2] = l2;
  out_loc[o + 3] = l3;
  out_loc[o + 4] = l4;
  out_conf[(size_t)b * kP + p] = (float)conf;
}

// ---------------------------------------------------------------------------
// Inputs (setup_inputs order): 0=localisations (unused), 1=confidences (unused),
// 2=priors (P*5 f32), 3=targets (B*T*6 f32).
// Output: matched_locs (B,P,5) then matched_confs (B,P), flat.
// ---------------------------------------------------------------------------
extern "C" void kernel_launch(void* const* d_in, const int* in_sizes, int n_in,
                              void* d_out, int out_size, void* d_ws, size_t ws_size,
                              hipStream_t stream) {
  (void)in_sizes; (void)n_in; (void)out_size; (void)ws_size;

  const float* priors  = (const float*)d_in[2];
  const float* targets = (const float*)d_in[3];
  float* out      = (float*)d_out;
  float* out_loc  = out;
  float* out_conf = out + (size_t)kB * kP * 5;
  unsigned long long* ws = (unsigned long long*)d_ws;  // kB*kT packed u64 (6.4 KB)

  ws_init_kernel<<<dim3((kB * kT + kBlk - 1) / kBlk), kBlk, 0, stream>>>(ws);
  best_prior_kernel<<<dim3(kP / (kBlk * kPRR), kB), kBlk, 0, stream>>>(priors, targets, ws);
  match_encode_kernel<<<dim3(kP / kBlk, kB), kBlk, 0, stream>>>(priors, targets, ws,
                                                                out_loc, out_conf);
}